// Synapse_71992241816165
// MI455X (gfx1250) — compile-verified
//
#include <hip/hip_runtime.h>
#include <hip/hip_bf16.h>

typedef __attribute__((ext_vector_type(2))) float v2f;
typedef __attribute__((ext_vector_type(8))) float v8f;

#define HDIM 256
#define NROW 4096
#define POWS 8
#define ROWBLOCKS 32
#define ROWS_PER_BLOCK (NROW / ROWBLOCKS)    // 128
#define ROWS_PER_WAVE  (ROWS_PER_BLOCK / 8)  // 16 (8 waves per 256-thread block)

// ---------------------------------------------------------------------------
// Stage 1: column power sums a_k[i] = sum_n x[n,i]^k, k=1..8.
// Each wave owns a 16-column stripe; VALU builds powers, WMMA (ones-A,
// f32 16x16x4) folds 8 accumulated row-partials per round into 8 f32
// accumulator tiles. With A==ones, C[m][j] = column-j sum in every row,
// so acc[k][0] of any lane holds the stripe sum for column (lane&15).
// ---------------------------------------------------------------------------
__global__ void __launch_bounds__(256) synapse_colpow_wmma(
    const float* __restrict__ x, float* __restrict__ part) {
  const int stripe = blockIdx.x & 15;   // 16 column stripes of 16
  const int rowBlk = blockIdx.x >> 4;   // ROWBLOCKS row blocks
  const int wave   = threadIdx.x >> 5;  // 8 waves per block
  const int lane   = threadIdx.x & 31;
  const int col    = (stripe << 4) | (lane & 15);
  const int half   = lane >> 4;         // which K-slot pair this lane feeds

  v2f ones; ones[0] = 1.0f; ones[1] = 1.0f;

  v8f acc[POWS];
#pragma unroll
  for (int k = 0; k < POWS; ++k)
#pragma unroll
    for (int e = 0; e < 8; ++e) acc[k][e] = 0.0f;

  const int row0 = rowBlk * ROWS_PER_BLOCK + wave * ROWS_PER_WAVE;
#pragma unroll
  for (int rc = 0; rc < ROWS_PER_WAVE / 8; ++rc) {
    const int r = row0 + rc * 8;
    v2f b[POWS];
#pragma unroll
    for (int it = 0; it < 2; ++it) {           // VALU pre-accumulate 8 rows
      const int rr = r + it * 4 + half * 2;
      const float t0 = x[rr * HDIM + col];
      const float t1 = x[(rr + 1) * HDIM + col];
      v2f t; t[0] = t0; t[1] = t1;
      v2f pk = t;
      if (it == 0) {
        b[0] = pk;
#pragma unroll
        for (int k = 1; k < POWS; ++k) { pk *= t; b[k] = pk; }
      } else {
        b[0] += pk;
#pragma unroll
        for (int k = 1; k < POWS; ++k) { pk *= t; b[k] += pk; }
      }
    }
    // Matrix pipe does the cross-row/cross-lane reduction for all 8 powers.
#pragma unroll
    for (int k = 0; k < POWS; ++k)
      acc[k] = __builtin_amdgcn_wmma_f32_16x16x4_f32(
          false, ones, false, b[k], (short)0, acc[k], false, false);
  }

  // Merge the 8 waves of this block (same stripe, disjoint rows) via LDS,
  // then emit one deterministic partial per (rowBlk, k, column).
  __shared__ float red[8][POWS][16];
  if (lane < 16) {
#pragma unroll
    for (int k = 0; k < POWS; ++k) red[wave][k][lane] = acc[k][0];
  }
  __syncthreads();
  if (threadIdx.x < POWS * 16) {
    const int k = threadIdx.x >> 4;
    const int j = threadIdx.x & 15;
    float s = 0.0f;
#pragma unroll
    for (int w = 0; w < 8; ++w) s += red[w][k][j];
    part[rowBlk * (POWS * HDIM) + k * HDIM + (stripe << 4) + j] = s;
  }
}

// ---------------------------------------------------------------------------
// Stage 1.5: fixed-order reduce of row-block partials -> a[8][256].
// ---------------------------------------------------------------------------
__global__ void __launch_bounds__(256) synapse_reduce_part(
    const float* __restrict__ part, float* __restrict__ a) {
  const int idx = blockIdx.x * 256 + threadIdx.x;  // 0..2047
  float s = 0.0f;
#pragma unroll
  for (int rb = 0; rb < ROWBLOCKS; ++rb) s += part[rb * (POWS * HDIM) + idx];
  a[idx] = s;
}

// ---------------------------------------------------------------------------
// Stage 2: E[o] = exp(-sum_i sum_k a_k[i] * W[o,i]^k / k). One block per o.
// ---------------------------------------------------------------------------
__global__ void __launch_bounds__(256) synapse_scale(
    const float* __restrict__ W, const float* __restrict__ a,
    float* __restrict__ E) {
  const int o = blockIdx.x;
  const int i = threadIdx.x;
  const float w  = W[o * HDIM + i];
  const float w2 = w * w,   w3 = w2 * w, w4 = w2 * w2;
  const float w5 = w4 * w,  w6 = w4 * w2, w7 = w4 * w3, w8 = w4 * w4;
  const float v =
      a[0 * HDIM + i] * w +
      0.5f          * a[1 * HDIM + i] * w2 +
      (1.0f / 3.0f) * a[2 * HDIM + i] * w3 +
      0.25f         * a[3 * HDIM + i] * w4 +
      0.2f          * a[4 * HDIM + i] * w5 +
      (1.0f / 6.0f) * a[5 * HDIM + i] * w6 +
      (1.0f / 7.0f) * a[6 * HDIM + i] * w7 +
      0.125f        * a[7 * HDIM + i] * w8;

  __shared__ float s[HDIM];
  s[i] = v;
  __syncthreads();
  for (int st = HDIM / 2; st > 0; st >>= 1) {
    if (i < st) s[i] += s[i + st];
    __syncthreads();
  }
  if (i == 0) E[o] = expf(-s[0]);   // exp(S[o]), S[o] = -s[0]
}

// ---------------------------------------------------------------------------
// Stage 3: out[n,o] = x[n,o] * E[o], float4-vectorized (bandwidth-bound).
// ---------------------------------------------------------------------------
__global__ void __launch_bounds__(256) synapse_out(
    const float* __restrict__ x, const float* __restrict__ E,
    float* __restrict__ out) {
  const int idx = blockIdx.x * 256 + threadIdx.x;          // float4 index
  const float4 xv = ((const float4*)x)[idx];
  const float4 ev = ((const float4*)E)[idx & (HDIM / 4 - 1)];
  float4 r;
  r.x = xv.x * ev.x; r.y = xv.y * ev.y;
  r.z = xv.z * ev.z; r.w = xv.w * ev.w;
  ((float4*)out)[idx] = r;
}

extern "C" void kernel_launch(void* const* d_in, const int* in_sizes, int n_in,
                              void* d_out, int out_size, void* d_ws, size_t ws_size,
                              hipStream_t stream) {
  (void)in_sizes; (void)n_in; (void)out_size; (void)ws_size;
  const float* x = (const float*)d_in[0];  // (4096, 256) f32
  const float* W = (const float*)d_in[1];  // (256, 256)  f32
  float* out = (float*)d_out;              // (4096, 256) f32

  float* part = (float*)d_ws;                      // [ROWBLOCKS][8][256] = 256 KB
  float* a    = part + ROWBLOCKS * POWS * HDIM;    // [8][256]
  float* E    = a + POWS * HDIM;                   // [256]

  synapse_colpow_wmma<<<dim3(16 * ROWBLOCKS), dim3(256), 0, stream>>>(x, part);
  synapse_reduce_part<<<dim3(POWS * HDIM / 256), dim3(256), 0, stream>>>(part, a);
  synapse_scale<<<dim3(HDIM), dim3(256), 0, stream>>>(W, a, E);
  synapse_out<<<dim3(NROW * HDIM / 4 / 256), dim3(256), 0, stream>>>(x, E, out);
}